// NMS_Export_17506286699228
// MI455X (gfx1250) — compile-verified
//
#include <hip/hip_runtime.h>
#include <stdint.h>

#define NANCH   25200
#define NCH     30
#define NCLS    4
#define TOPK    1024
#define MAXDET  300
#define NSORT   32768
#define RPB     256       // anchors per score block
#define CONF_T  0.001f
#define IOU_T   0.45f
#define CLS_OFF 4096.0f
#define OUTC    27        // 4 box + 1 sc + 1 cls + 21 kpt
#define ROWW    33        // padded words per suppression row (bank-conflict avoidance)

typedef unsigned int u32x4 __attribute__((ext_vector_type(4)));
typedef int          i32x8 __attribute__((ext_vector_type(8)));
typedef int          i32x4 __attribute__((ext_vector_type(4)));

// ---------------------------------------------------------------------------
// Kernel 1: per-anchor score. Stage 256x30 f32 tile into LDS via the Tensor
// Data Mover (CDNA5 TDM: tensor_load_to_lds + s_wait_tensorcnt), then compute
// score = (obj>T && conf>T) ? conf : -1, conf = max(cls[0..3]*obj).
// ---------------------------------------------------------------------------
__global__ __launch_bounds__(RPB) void score_kernel(const float* __restrict__ x,
                                                    float* __restrict__ scores) {
#pragma clang fp contract(off)
  __shared__ float tile[RPB * NCH];
  const int b  = blockIdx.y;
  const int a0 = blockIdx.x * RPB;
  const int n  = (NANCH - a0 < RPB) ? (NANCH - a0) : RPB;
  const int elems = n * NCH;
  const float* src = x + ((size_t)b * NANCH + a0) * NCH;

#if defined(__HIP_DEVICE_COMPILE__)
  if (threadIdx.x < 32) {   // one wave issues the TDM transfer (EXEC ignored by TDM)
    uint64_t ga  = (uint64_t)(uintptr_t)src;
    uint32_t lds = (uint32_t)(uintptr_t)&tile[0];   // aperture low-32 = LDS byte offset
    u32x4 g0;
    g0[0] = 1u;                                             // count=1 descriptor
    g0[1] = lds;                                            // lds_addr
    g0[2] = (uint32_t)ga;                                   // global_addr[31:0]
    g0[3] = ((uint32_t)(ga >> 32) & 0x01FFFFFFu) | (2u << 30); // addr[56:32] | type=2
    i32x8 g1;
    g1[0] = (int)(2u << 16);                                // data_size = 4B
    g1[1] = (int)(((uint32_t)elems & 0xFFFFu) << 16);       // tensor_dim0[15:0]
    g1[2] = (int)(((uint32_t)elems >> 16) | (1u << 16));    // tensor_dim0[31:16], tensor_dim1=1
    g1[3] = (int)((uint32_t)elems << 16);                   // tile_dim0 = elems (<=7680)
    g1[4] = 1;                                              // tile_dim1 = 1
    g1[5] = (int)elems;                                     // tensor_dim0_stride
    g1[6] = 0; g1[7] = 0;
    i32x4 z4 = {0, 0, 0, 0};
#if __clang_major__ >= 23
    i32x8 z8 = {0, 0, 0, 0, 0, 0, 0, 0};
    __builtin_amdgcn_tensor_load_to_lds(g0, g1, z4, z4, z8, 0);
#else
    __builtin_amdgcn_tensor_load_to_lds(g0, g1, z4, z4, 0);
#endif
    __builtin_amdgcn_s_wait_tensorcnt(0);
  }
  __syncthreads();
#else
  for (int i = threadIdx.x; i < elems; i += blockDim.x) tile[i] = src[i];
  __syncthreads();
#endif

  const int t = threadIdx.x;
  if (t < n) {
    const float* r = &tile[t * NCH];
    float obj  = r[4];
    float best = r[5] * obj;
#pragma unroll
    for (int c = 1; c < NCLS; ++c) {
      float v = r[5 + c] * obj;
      if (v > best) best = v;
    }
    bool valid = (obj > CONF_T) && (best > CONF_T);
    scores[(size_t)b * NANCH + a0 + t] = valid ? best : -1.0f;
  }
}

// ---------------------------------------------------------------------------
// Kernel 2: exact top-1024 per batch via full bitonic sort of 32768 packed
// 64-bit keys in 256KB dynamic LDS (CDNA5 WGP: 320KB LDS).
// key = (monotone(score) << 32) | ~index  -> descending sort == jax top_k order.
// ---------------------------------------------------------------------------
__global__ __launch_bounds__(1024) void sort_kernel(const float* __restrict__ scores,
                                                    float* __restrict__ sc_out,
                                                    int* __restrict__ idx_out) {
  extern __shared__ unsigned long long keys[];
  const int b   = blockIdx.x;
  const int tid = threadIdx.x;
  const float* s = scores + (size_t)b * NANCH;

  for (int i = tid; i < NSORT; i += 1024) {
    unsigned long long k = 0ull;   // padding sorts last (all real keys > 0)
    if (i < NANCH) {
      unsigned sb = __float_as_uint(s[i]);
      unsigned u  = (sb & 0x80000000u) ? ~sb : (sb | 0x80000000u);
      k = ((unsigned long long)u << 32) | (unsigned)(~(unsigned)i);
    }
    keys[i] = k;
  }
  __syncthreads();

  for (unsigned K = 2; K <= NSORT; K <<= 1) {
    for (unsigned J = K >> 1; J > 0; J >>= 1) {
      for (unsigned t = tid; t < NSORT / 2; t += 1024) {
        unsigned i = ((t & ~(J - 1u)) << 1) | (t & (J - 1u));
        unsigned p = i | J;
        unsigned long long a = keys[i], c = keys[p];
        bool up = (i & K) == 0;                 // descending network
        bool sw = up ? (a < c) : (a > c);
        if (sw) { keys[i] = c; keys[p] = a; }
      }
      __syncthreads();
    }
  }

  {
    unsigned long long k = keys[tid];           // tid in [0,1024)
    unsigned u  = (unsigned)(k >> 32);
    unsigned sb = (u & 0x80000000u) ? (u ^ 0x80000000u) : ~u;
    sc_out[b * TOPK + tid]  = __uint_as_float(sb);
    idx_out[b * TOPK + tid] = (int)(~(unsigned)k);
  }
}

// ---------------------------------------------------------------------------
// Kernel 3: greedy NMS via bit-matrix + single-wave serial scan, then stable
// compaction (== reference's zero-mask + second top_k).
// ---------------------------------------------------------------------------
__global__ __launch_bounds__(1024) void nms_kernel(const float* __restrict__ x,
                                                   const float* __restrict__ sc_in,
                                                   const int* __restrict__ idx_in,
                                                   float* __restrict__ out) {
#pragma clang fp contract(off)
  extern __shared__ char smem[];
  float4*   sbox  = (float4*)smem;                            // 16 KB
  unsigned* rows  = (unsigned*)(smem + 16384);                // 1024*33*4 = 132 KB
  unsigned* keepw = (unsigned*)(smem + 16384 + TOPK * ROWW * 4); // 32 words
  unsigned* pref  = keepw + 32;                               // 33 words

  const int b = blockIdx.x;
  const int j = threadIdx.x;
  const int   idx = idx_in[b * TOPK + j];
  const float sc  = sc_in[b * TOPK + j];
  const float* p  = x + ((size_t)b * NANCH + idx) * NCH;
  __builtin_prefetch(p, 0, 0);

  float cx = p[0], cy = p[1], w = p[2], h = p[3], obj = p[4];
  float best = p[5] * obj; int cls = 0;
#pragma unroll
  for (int c = 1; c < NCLS; ++c) {
    float v = p[5 + c] * obj;
    if (v > best) { best = v; cls = c; }
  }
  float bx1 = cx - w * 0.5f, by1 = cy - h * 0.5f;
  float bx2 = cx + w * 0.5f, by2 = cy + h * 0.5f;
  float off = (float)cls * CLS_OFF;
  float x1 = bx1 + off, y1 = by1 + off, x2 = bx2 + off, y2 = by2 + off;
  sbox[j] = make_float4(x1, y1, x2, y2);

  unsigned long long ball = __ballot(sc > 0.0f);
  if ((j & 31) == 0) keepw[j >> 5] = (unsigned)ball;
  __syncthreads();

  // Each thread builds suppression row j: bit jj set iff iou(j,jj)>T && jj>j
  float areai = (x2 - x1) * (y2 - y1);
  unsigned word = 0;
  for (int jj = 0; jj < TOPK; ++jj) {
    float4 bb = sbox[jj];                       // broadcast read
    float lx = fmaxf(x1, bb.x), ly = fmaxf(y1, bb.y);
    float rx = fminf(x2, bb.z), ry = fminf(y2, bb.w);
    float iw = fmaxf(rx - lx, 0.0f), ih = fmaxf(ry - ly, 0.0f);
    float inter = iw * ih;
    float areaj = (bb.z - bb.x) * (bb.w - bb.y);
    float iou = inter / (areai + areaj - inter + 1e-9f);
    if (iou > IOU_T && jj > j) word |= (1u << (jj & 31));
    if ((jj & 31) == 31) { rows[j * ROWW + (jj >> 5)] = word; word = 0; }
  }
  __syncthreads();

  // Serial greedy scan, one wave32, lane L owns keep bits [32L, 32L+32)
  if (j < 32) {
    unsigned kw = keepw[j];
    for (int i = 0; i < TOPK; ++i) {
      unsigned wv = (unsigned)__shfl((int)kw, i >> 5, 32);
      if ((wv >> (i & 31)) & 1u) kw &= ~rows[i * ROWW + j];
    }
    keepw[j] = kw;
  }
  __syncthreads();

  if (j == 0) {
    unsigned run = 0;
    for (int wdx = 0; wdx < 32; ++wdx) { pref[wdx] = run; run += __popc(keepw[wdx]); }
    pref[32] = run;
  }
  __syncthreads();

  const int nkeep = (int)pref[32];
  unsigned myw = keepw[j >> 5];
  bool kept = (myw >> (j & 31)) & 1u;
  int rank = (int)pref[j >> 5] + __popc(myw & ((1u << (j & 31)) - 1u));
  float* ob = out + (size_t)b * MAXDET * OUTC;

  if (kept && rank < MAXDET) {
    float* o = ob + rank * OUTC;
    o[0] = bx1; o[1] = by1; o[2] = bx2; o[3] = by2;
    o[4] = sc;  o[5] = (float)cls;
#pragma unroll
    for (int k = 0; k < 21; ++k) o[6 + k] = p[9 + k];
  }
  if (j < MAXDET && j >= nkeep) {
    float* o = ob + j * OUTC;
#pragma unroll
    for (int k = 0; k < OUTC; ++k) o[k] = 0.0f;
  }
}

// ---------------------------------------------------------------------------
extern "C" void kernel_launch(void* const* d_in, const int* in_sizes, int n_in,
                              void* d_out, int out_size, void* d_ws, size_t ws_size,
                              hipStream_t stream) {
  (void)in_sizes; (void)n_in; (void)out_size; (void)ws_size;
  const float* x = (const float*)d_in[0];
  float* out = (float*)d_out;

  // workspace layout (floats): scores[32*25200] | sc[32*1024] | idx[32*1024]
  float* wsf    = (float*)d_ws;
  float* scores = wsf;
  float* sc_ws  = wsf + (size_t)32 * NANCH;
  int*   idx_ws = (int*)(sc_ws + (size_t)32 * TOPK);

  dim3 gs((NANCH + RPB - 1) / RPB, 32);
  score_kernel<<<gs, RPB, 0, stream>>>(x, scores);

  sort_kernel<<<32, 1024, (size_t)NSORT * sizeof(unsigned long long), stream>>>(
      scores, sc_ws, idx_ws);

  const size_t nms_smem = 16384 + (size_t)TOPK * ROWW * 4 + 65 * 4 + 256;
  nms_kernel<<<32, 1024, nms_smem, stream>>>(x, sc_ws, idx_ws, out);
}